// GNNFakeNews_37520834298467
// MI455X (gfx1250) — compile-verified
//
#include <hip/hip_runtime.h>

#define N_NODES 100000
#define N_EDGES 1250000
#define IN_CH 64
#define HID 128
#define OUT_CH 2
#define NUM_GRAPHS 128

// B stored K-pair interleaved: Bs[p][2n+j] = B[2p+j][n], row stride 288 floats.
// 288 % 64 == 32 -> lanes 0-15 (pair p) and 16-31 (pair p+1) hit disjoint banks.
#define BS_STRIDE 288
// A[r][k] rows padded: 132 % 64 == 4 -> conflict-free fragment reads.
#define AS_STRIDE 132

typedef float v2f __attribute__((ext_vector_type(2)));
typedef float v8f __attribute__((ext_vector_type(8)));

// ---------------------------------------------------------------------------
// Kernel 1: edge scatter (mean-aggregation numerator + degree count).
// One wave32 per edge; lane handles channels 2*lane, 2*lane+1.
// Grid-stride over edges with a software prefetch of the next gather row.
// ---------------------------------------------------------------------------
__global__ __launch_bounds__(256) void edge_scatter_kernel(
    const float* __restrict__ x, const long long* __restrict__ ei,
    float* __restrict__ agg, float* __restrict__ cnt, int nWaves) {
  const int wid  = (blockIdx.x * blockDim.x + threadIdx.x) >> 5;
  const int lane = threadIdx.x & 31;
  for (int e = wid; e < N_EDGES; e += nWaves) {
    long long s = ei[e];              // src (uniform per wave -> scalarized)
    long long d = ei[N_EDGES + e];    // dst
    int en = e + nWaves;
    if (en < N_EDGES) {
      long long sn = ei[en];
      __builtin_prefetch(x + sn * IN_CH + lane * 2, 0, 0);  // global_prefetch_b8
    }
    const float* xs = x + s * IN_CH + (long long)lane * 2;
    float v0 = xs[0];
    float v1 = xs[1];
    float* dp = agg + d * IN_CH + (long long)lane * 2;
    atomicAdd(dp, v0);                // global_atomic_add_f32
    atomicAdd(dp + 1, v1);
    if (lane == 0) atomicAdd(cnt + d, 1.0f);
  }
}

// ---------------------------------------------------------------------------
// Kernel 2: fused node GEMM + relu + per-graph max-pool.
//   h[node, :] = relu([mean|x] @ [W_l;W_r]^T + b_l)   (K = 128)
// 256 threads = 8 waves; each wave owns a 16-column strip of a 16x128 tile.
// Weights live in LDS K-pair interleaved so each B fragment is one
// ds_load_b64; A tile rebuilt per 16 nodes, fragments are ds_load_b64 too.
// Pooling fused via integer atomicMax on relu'd (non-negative) f32 bits.
// ---------------------------------------------------------------------------
__global__ __launch_bounds__(256) void node_wmma_kernel(
    const float* __restrict__ x, const float* __restrict__ agg,
    const float* __restrict__ cnt, const float* __restrict__ W_l,
    const float* __restrict__ b_l, const float* __restrict__ W_r,
    const long long* __restrict__ batch, float* __restrict__ g) {
  extern __shared__ float smem[];
  float* Bs = smem;                        // [64][BS_STRIDE] K-pair interleaved
  float* As = Bs + (HID / 2) * BS_STRIDE;  // [16][AS_STRIDE] : A[r][k] = {mean|x}
  float* bl = As + 16 * AS_STRIDE;         // [128]

  // Stage Wcat^T once per block, K-pair interleaved.
  // i -> (p = K-pair 0..63, n = output channel 0..127); adjacent float2 LDS
  // stores (conflict-free); global reads are 8B @ 256B stride, L2-resident.
  for (int i = threadIdx.x; i < (HID / 2) * HID; i += 256) {
    int p = i >> 7;                        // K-pair index
    int n = i & 127;                       // output channel
    const float* W = (p < IN_CH / 2) ? W_l : W_r;
    int kk = (p & (IN_CH / 2 - 1)) << 1;   // k within W_l / W_r
    v2f w;
    w.x = W[n * IN_CH + kk];
    w.y = W[n * IN_CH + kk + 1];
    *(v2f*)(Bs + p * BS_STRIDE + 2 * n) = w;
  }
  if (threadIdx.x < HID) bl[threadIdx.x] = b_l[threadIdx.x];
  __syncthreads();

  const int wave  = threadIdx.x >> 5;
  const int lane  = threadIdx.x & 31;
  const int mrow  = lane & 15;
  const int khalf = (lane >> 4) << 1;      // 0 for lanes 0-15, 2 for lanes 16-31
  const int n0    = wave << 4;             // this wave's column strip
  const int numTiles = (N_NODES + 15) / 16;

  // Per-lane base pointers for the fragment loads.
  const float* aBase = As + mrow * AS_STRIDE + khalf;          // + k
  const float* bBase = Bs + (khalf >> 1) * BS_STRIDE + 2 * (n0 + mrow); // + (k/2)*STR

  for (int tile = blockIdx.x; tile < numTiles; tile += gridDim.x) {
    const int row0 = tile << 4;

    // Build A tile: cols 0..63 = mean = agg/max(cnt,1), cols 64..127 = x.
    for (int i = threadIdx.x; i < 16 * IN_CH; i += 256) {
      int r = i >> 6;
      int k = i & 63;
      int node = row0 + r;
      float m = 0.0f, xv = 0.0f;
      if (node < N_NODES) {
        float c = cnt[node];
        c = c > 1.0f ? c : 1.0f;
        m  = agg[(long long)node * IN_CH + k] / c;
        xv = x[(long long)node * IN_CH + k];
      }
      As[r * AS_STRIDE + k]         = m;
      As[r * AS_STRIDE + k + IN_CH] = xv;
    }
    __syncthreads();

    // 16x16 output tile per wave, K=128 in steps of 4 (32 WMMAs).
    v8f acc = {};
#pragma unroll
    for (int k = 0; k < HID; k += 4) {
      v2f a = *(const v2f*)(aBase + k);                        // A[M][K..K+1]
      v2f b = *(const v2f*)(bBase + (k >> 1) * BS_STRIDE);     // B[K..K+1][N]
      acc = __builtin_amdgcn_wmma_f32_16x16x4_f32(
          /*neg_a=*/false, a, /*neg_b=*/false, b,
          /*c_mod=*/(short)0, acc, /*reuse_a=*/false, /*reuse_b=*/false);
    }

    // bias + relu + fused segment_max (int-compare valid for x >= 0).
    const int   colg  = n0 + mrow;
    const float bias  = bl[colg];
    const int   mbase = (lane >> 4) << 3;  // D vgpr v -> row v + mbase
#pragma unroll
    for (int v = 0; v < 8; v++) {
      int node = row0 + mbase + v;
      if (node < N_NODES) {
        float h = acc[v] + bias;
        h = h > 0.0f ? h : 0.0f;
        int gi = (int)batch[node];
        atomicMax((int*)(g + gi * HID + colg), __float_as_int(h));
      }
    }
    __syncthreads();
  }
}

// ---------------------------------------------------------------------------
// Kernel 3: tiny head, out[G,2] = g @ W_lin^T + b_lin  (65 KFLOP, scalar).
// ---------------------------------------------------------------------------
__global__ __launch_bounds__(256) void head_kernel(
    const float* __restrict__ g, const float* __restrict__ W_lin,
    const float* __restrict__ b_lin, float* __restrict__ out) {
  int i = threadIdx.x;
  if (i < NUM_GRAPHS * OUT_CH) {
    int gi = i >> 1, o = i & 1;
    float s = b_lin[o];
    for (int k = 0; k < HID; k++) s += g[gi * HID + k] * W_lin[o * HID + k];
    out[i] = s;
  }
}

extern "C" void kernel_launch(void* const* d_in, const int* in_sizes, int n_in,
                              void* d_out, int out_size, void* d_ws, size_t ws_size,
                              hipStream_t stream) {
  const float*     x      = (const float*)d_in[0];
  const float*     W_l    = (const float*)d_in[1];
  const float*     b_l    = (const float*)d_in[2];
  const float*     W_r    = (const float*)d_in[3];
  const float*     W_lin  = (const float*)d_in[4];
  const float*     b_lin  = (const float*)d_in[5];
  const long long* ei     = (const long long*)d_in[6];   // int64 [2, E]
  const long long* batch  = (const long long*)d_in[7];   // int64 [N]
  float*           out    = (float*)d_out;

  // Workspace layout: agg[N*64] | cnt[N] | g[128*128]   (~26.1 MB)
  float* agg = (float*)d_ws;
  float* cnt = agg + (size_t)N_NODES * IN_CH;
  float* g   = cnt + N_NODES;
  size_t zeroBytes =
      ((size_t)N_NODES * IN_CH + N_NODES + (size_t)NUM_GRAPHS * HID) * sizeof(float);
  hipMemsetAsync(d_ws, 0, zeroBytes, stream);

  // Edge scatter: 8192 blocks * 8 waves = 64K waves, ~19 edges each.
  const int edgeBlocks = 8192;
  const int nWaves = edgeBlocks * (256 / 32);
  edge_scatter_kernel<<<edgeBlocks, 256, 0, stream>>>(x, ei, agg, cnt, nWaves);

  // Node GEMM: 2048 blocks grid-stride over 6250 row tiles, ~82.7 KB dyn LDS.
  const size_t shmem =
      ((HID / 2) * BS_STRIDE + 16 * AS_STRIDE + HID) * sizeof(float);
  node_wmma_kernel<<<2048, 256, shmem, stream>>>(x, agg, cnt, W_l, b_l, W_r,
                                                 batch, g);

  head_kernel<<<1, 256, 0, stream>>>(g, W_lin, b_lin, out);
}